// UnsupervisedGAT_18468359373276
// MI455X (gfx1250) — compile-verified
//
#include <hip/hip_runtime.h>
#include <hip/hip_bf16.h>
#include <cstdint>
#include <cstddef>

// ---------------------------------------------------------------------------
// UnsupervisedGAT (3x GATConv) for MI455X / gfx1250, wave32.
// Dense projections use V_WMMA_F32_16X16X4_F32 (exact fp32 matrix path).
// Sparse softmax/aggregation use ordered-uint atomicMax + f32 atomicAdd.
// ---------------------------------------------------------------------------

typedef __attribute__((ext_vector_type(2))) float v2f;
typedef __attribute__((ext_vector_type(8))) float v8f;

#define TPB 256

// Ordered-uint encoding for float atomic max (monotone: enc(a)<enc(b) <=> a<b)
__device__ __forceinline__ unsigned enc_ord(float f) {
  unsigned u = __float_as_uint(f);
  return (u & 0x80000000u) ? ~u : (u | 0x80000000u);
}
__device__ __forceinline__ float dec_ord(unsigned u) {
  return __uint_as_float((u & 0x80000000u) ? (u ^ 0x80000000u) : ~u);
}
#define ENC_NEG_INF 0x007FFFFFu  // enc_ord(-inf)

// ---------------------------------------------------------------------------
// h[N x 64] = x[N x K] @ W[K x 64], K in {128, 64}. fp32 WMMA.
// Block: 256 threads = 8 waves -> 2 row-tiles x 4 col-tiles = 32 x 64 output.
// LDS: X tile (32 x K) + full W (K x 64)  (<= 48 KB).
// ---------------------------------------------------------------------------
__global__ __launch_bounds__(TPB) void gat_gemm_wmma(
    const float* __restrict__ x, const float* __restrict__ W,
    float* __restrict__ h, int N, int K)
{
  __shared__ float sX[32 * 128];
  __shared__ float sW[128 * 64];

  const int tid = threadIdx.x;
  const int rowblk = blockIdx.x * 32;

  // Cooperative coalesced load of W (K*64 floats)
  for (int i = tid; i < K * 64; i += TPB) sW[i] = W[i];
  // Cooperative load of X tile (32*K floats), clamp OOB rows (dup reads only)
  for (int i = tid; i < 32 * K; i += TPB) {
    int r = i / K, c = i - r * K;
    int gr = rowblk + r; if (gr >= N) gr = N - 1;
    sX[r * K + c] = x[(size_t)gr * K + c];
  }
  __syncthreads();

  const int wave   = tid >> 5;
  const int lane   = tid & 31;
  const int half   = lane >> 4;       // lane-half selects K sub-pair
  const int l16    = lane & 15;
  const int rowTile = (wave >> 2) * 16;   // 0 or 16
  const int colBase = (wave & 3) * 16;    // 0,16,32,48

  v8f acc = {};
  for (int k0 = 0; k0 < K; k0 += 4) {
    const int ka = k0 + 2 * half;   // K offset for this lane-half (VGPR0/1 pair)
    v2f a, b;
    // A 16x4 f32: lanes 0-15 hold K={0,1}, lanes 16-31 hold K={2,3}
    const float2 av = *(const float2*)&sX[(rowTile + l16) * K + ka];
    a.x = av.x; a.y = av.y;
    // B 4x16 f32: same K split across lane-halves, N striped over lanes
    b.x = sW[(ka    ) * 64 + colBase + l16];
    b.y = sW[(ka + 1) * 64 + colBase + l16];
    acc = __builtin_amdgcn_wmma_f32_16x16x4_f32(
        false, a, false, b, (short)0, acc, false, false);
  }

  // C/D 16x16 f32: VGPR v holds row v (lanes 0-15) / row v+8 (lanes 16-31)
  #pragma unroll
  for (int v = 0; v < 8; ++v) {
    int gr = rowblk + rowTile + v + 8 * half;
    if (gr < N) h[(size_t)gr * 64 + colBase + l16] = acc[v];
  }
}

// ---------------------------------------------------------------------------
// el/er[N,H]: per-(node,head) dot of h[n,h,:] with attn_l/attn_r[h,:]
// ---------------------------------------------------------------------------
__global__ __launch_bounds__(TPB) void gat_elr(
    const float* __restrict__ h, const float* __restrict__ attn_l,
    const float* __restrict__ attn_r, float* __restrict__ el,
    float* __restrict__ er, int NH)
{
  int i = blockIdx.x * TPB + threadIdx.x;
  if (i >= NH) return;
  int n = i >> 3, hh = i & 7;
  const float* hp = h + (size_t)n * 64 + hh * 8;
  const float* al = attn_l + hh * 8;
  const float* ar = attn_r + hh * 8;
  float sl = 0.f, sr = 0.f;
  #pragma unroll
  for (int d = 0; d < 8; ++d) { float v = hp[d]; sl += v * al[d]; sr += v * ar[d]; }
  el[i] = sl; er[i] = sr;
}

__global__ __launch_bounds__(TPB) void gat_init_ms(
    unsigned* __restrict__ menc, float* __restrict__ s, int NH)
{
  int i = blockIdx.x * TPB + threadIdx.x;
  if (i >= NH) return;
  menc[i] = ENC_NEG_INF;
  s[i] = 0.f;
}

__global__ __launch_bounds__(TPB) void gat_zero(float* __restrict__ p, int n)
{
  int i = blockIdx.x * TPB + threadIdx.x;
  if (i < n) p[i] = 0.f;
}

// e = leaky_relu(el[src]+er[dst], 0.2); segment max over dst via ordered atomicMax
__global__ __launch_bounds__(TPB) void gat_edge_logits(
    const float* __restrict__ el, const float* __restrict__ er,
    const int* __restrict__ src, const int* __restrict__ dst,
    float* __restrict__ exb, unsigned* __restrict__ menc, int EH)
{
  int i = blockIdx.x * TPB + threadIdx.x;
  if (i >= EH) return;
  int e = i >> 3, hh = i & 7;
  float v = el[src[e] * 8 + hh] + er[dst[e] * 8 + hh];
  v = (v > 0.f) ? v : 0.2f * v;
  exb[i] = v;
  atomicMax(&menc[dst[e] * 8 + hh], enc_ord(v));
}

// ex = exp(e - m[dst]); segment sum via atomicAdd
__global__ __launch_bounds__(TPB) void gat_edge_exp(
    float* __restrict__ exb, const unsigned* __restrict__ menc,
    float* __restrict__ s, const int* __restrict__ dst, int EH)
{
  int i = blockIdx.x * TPB + threadIdx.x;
  if (i >= EH) return;
  int e = i >> 3, hh = i & 7;
  float m = dec_ord(menc[dst[e] * 8 + hh]);
  float ex = expf(exb[i] - m);
  exb[i] = ex;
  atomicAdd(&s[dst[e] * 8 + hh], ex);
}

// out[dst, c] += (ex/s[dst]) * h[src, c] ; 64 consecutive lanes share one edge
// -> h[src] gather is two 128B lines per edge; src/dst loads coalesce to one
//    request per wave. h (25.6 MB) fits in the 192 MB L2.
__global__ __launch_bounds__(TPB) void gat_aggregate(
    const float* __restrict__ exb, const float* __restrict__ s,
    const float* __restrict__ h, const int* __restrict__ src,
    const int* __restrict__ dst, float* __restrict__ out, int EC)
{
  int i = blockIdx.x * TPB + threadIdx.x;
  if (i >= EC) return;
  int e = i >> 6, c = i & 63, hh = c >> 3;
  int d0 = dst[e];
  float a = exb[e * 8 + hh] / s[d0 * 8 + hh];
  atomicAdd(&out[(size_t)d0 * 64 + c], a * h[(size_t)src[e] * 64 + c]);
}

__global__ __launch_bounds__(TPB) void gat_bias_act(
    float* __restrict__ out, const float* __restrict__ bias, int n, int act)
{
  int i = blockIdx.x * TPB + threadIdx.x;
  if (i >= n) return;
  float v = out[i] + bias[i & 63];
  if (act) v = (v > 0.f) ? v : 0.01f * v;
  out[i] = v;
}

// ---------------------------------------------------------------------------
static void run_layer(const float* x, const float* W, const float* al,
                      const float* ar, const float* bias, const int* src,
                      const int* dst, float* h, float* el, float* er,
                      unsigned* menc, float* s, float* exb, float* out,
                      int N, int E, int K, int act, hipStream_t stream)
{
  const int NH = N * 8, EH = E * 8, NC = N * 64, EC = E * 64;
  gat_gemm_wmma<<<dim3((N + 31) / 32), dim3(TPB), 0, stream>>>(x, W, h, N, K);
  gat_elr<<<dim3((NH + TPB - 1) / TPB), dim3(TPB), 0, stream>>>(h, al, ar, el, er, NH);
  gat_init_ms<<<dim3((NH + TPB - 1) / TPB), dim3(TPB), 0, stream>>>(menc, s, NH);
  gat_zero<<<dim3((NC + TPB - 1) / TPB), dim3(TPB), 0, stream>>>(out, NC);
  gat_edge_logits<<<dim3((EH + TPB - 1) / TPB), dim3(TPB), 0, stream>>>(el, er, src, dst, exb, menc, EH);
  gat_edge_exp<<<dim3((EH + TPB - 1) / TPB), dim3(TPB), 0, stream>>>(exb, menc, s, dst, EH);
  gat_aggregate<<<dim3((EC + TPB - 1) / TPB), dim3(TPB), 0, stream>>>(exb, s, h, src, dst, out, EC);
  gat_bias_act<<<dim3((NC + TPB - 1) / TPB), dim3(TPB), 0, stream>>>(out, bias, NC, act);
}

extern "C" void kernel_launch(void* const* d_in, const int* in_sizes, int n_in,
                              void* d_out, int out_size, void* d_ws, size_t ws_size,
                              hipStream_t stream)
{
  const float* n_feat = (const float*)d_in[0];
  // d_in[1] = e_feat (unused by GATConv)
  const int* src = (const int*)d_in[2];
  const int* dst = (const int*)d_in[3];
  const float* W0 = (const float*)d_in[4];
  const float* al0 = (const float*)d_in[5];
  const float* ar0 = (const float*)d_in[6];
  const float* b0 = (const float*)d_in[7];
  const float* W1 = (const float*)d_in[8];
  const float* al1 = (const float*)d_in[9];
  const float* ar1 = (const float*)d_in[10];
  const float* b1 = (const float*)d_in[11];
  const float* W2 = (const float*)d_in[12];
  const float* al2 = (const float*)d_in[13];
  const float* ar2 = (const float*)d_in[14];
  const float* b2 = (const float*)d_in[15];

  const int IN = 128;
  const int N = in_sizes[0] / IN;
  const int E = in_sizes[2];

  // Workspace layout (floats/uints), ~90 MB total:
  float* ws = (float*)d_ws;
  float* buf_h = ws;                                  // N*64
  float* buf_x = buf_h + (size_t)N * 64;              // N*64
  float* el    = buf_x + (size_t)N * 64;              // N*8
  float* er    = el + (size_t)N * 8;                  // N*8
  unsigned* menc = (unsigned*)(er + (size_t)N * 8);   // N*8
  float* s     = (float*)(menc + (size_t)N * 8);      // N*8
  float* exb   = s + (size_t)N * 8;                   // E*8

  // Layer 0: x = n_feat (K=128), act=leaky(0.01), out -> buf_x
  run_layer(n_feat, W0, al0, ar0, b0, src, dst,
            buf_h, el, er, menc, s, exb, buf_x, N, E, 128, 1, stream);
  // Layer 1: x = buf_x (K=64); gemm consumes buf_x before it is re-zeroed,
  // so aggregation can safely reuse it as the output buffer.
  run_layer(buf_x, W1, al1, ar1, b1, src, dst,
            buf_h, el, er, menc, s, exb, buf_x, N, E, 64, 1, stream);
  // Layer 2: x = buf_x (K=64), no activation, out -> d_out
  run_layer(buf_x, W2, al2, ar2, b2, src, dst,
            buf_h, el, er, menc, s, exb, (float*)d_out, N, E, 64, 0, stream);
}